// GIN_352187319043
// MI455X (gfx1250) — compile-verified
//
#include <hip/hip_runtime.h>
#include <hip/hip_bf16.h>

// ---------------------------------------------------------------------------
// GIN / GINE forward for MI455X (gfx1250, wave32).
// All linear layers use V_WMMA_F32_16X16X4_F32; each wave computes a 16x64
// output strip so every activation element is loaded exactly once per layer.
// ---------------------------------------------------------------------------

typedef float v2f __attribute__((ext_vector_type(2)));
typedef float v8f __attribute__((ext_vector_type(8)));

#define N_NODES 100000
#define N_EDGES 1600000
#define N_GRAPHS 64
#define IN_CH 16
#define HID 64
#define OUT_CH 4
#define BN_EPS 1e-5f

__device__ __forceinline__ void atomAddF32(float* p, float v) {
  // lowers to global_atomic_add_f32 (no return)
  unsafeAtomicAdd(p, v);
}

// -------------------------------- utility ----------------------------------
__global__ void k_zero(float* __restrict__ p, long n) {
  long i = (long)blockIdx.x * blockDim.x + threadIdx.x;
  if (i < n) p[i] = 0.0f;
}

// ------------------------- fused edge message+scatter -----------------------
// msg = relu(h[src] + edge_attr @ We + be); agg[dst] += msg
// grid: (ceil(E/256), C/4), block 256. One thread = one edge x 4 channels.
__global__ void k_edge_scatter(const float* __restrict__ h,
                               const int* __restrict__ src,
                               const int* __restrict__ dst,
                               const float* __restrict__ ea,
                               const float* __restrict__ We,
                               const float* __restrict__ be,
                               float* __restrict__ agg, int E, int C) {
  int e = blockIdx.x * blockDim.x + threadIdx.x;
  if (e >= E) return;
  int c0 = blockIdx.y << 2;
  int s = src[e], d = dst[e];
  float a0 = ea[2 * e + 0];
  float a1 = ea[2 * e + 1];
  // We is [2, C] row-major; uniform per block -> scalar loads
  float4 w0 = *reinterpret_cast<const float4*>(We + c0);
  float4 w1 = *reinterpret_cast<const float4*>(We + C + c0);
  float4 bb = *reinterpret_cast<const float4*>(be + c0);
  float4 hv = *reinterpret_cast<const float4*>(h + (size_t)s * C + c0);
  float m0 = fmaxf(hv.x + a0 * w0.x + a1 * w1.x + bb.x, 0.0f);
  float m1 = fmaxf(hv.y + a0 * w0.y + a1 * w1.y + bb.y, 0.0f);
  float m2 = fmaxf(hv.z + a0 * w0.z + a1 * w1.z + bb.z, 0.0f);
  float m3 = fmaxf(hv.w + a0 * w0.w + a1 * w1.w + bb.w, 0.0f);
  float* ap = agg + (size_t)d * C + c0;
  atomAddF32(ap + 0, m0);
  atomAddF32(ap + 1, m1);
  atomAddF32(ap + 2, m2);
  atomAddF32(ap + 3, m3);
}

// --------------------------- WMMA f32 GEMM (C=64) ---------------------------
// Y[N,64] = X[N,K] @ W[K,64] + bias.  N%16==0, K%4==0.
// block = (32,4): 4 waves, each wave computes one 16x64 strip of Y:
// one A fragment per K-step feeds 4 back-to-back V_WMMA_F32_16X16X4_F32.
// A 16x4 f32 frag (2 VGPR): lane l<16 holds M=l,K={k,k+1}; lane l+16 K={k+2,k+3}.
// B 4x16 frag mirrored: lane l<16 holds K={k,k+1},N=n0+l; lane l+16 K={k+2,k+3}.
// C/D 16x16 f32 (8 VGPR): vgpr r, lanes 0-15 -> M=r, lanes 16-31 -> M=r+8.
__global__ void k_wmma_gemm64(const float* __restrict__ X,
                              const float* __restrict__ W,
                              const float* __restrict__ bias,
                              float* __restrict__ Y, int N, int K) {
  const int C = 64;
  int lane = threadIdx.x;                      // 0..31
  int m0 = (blockIdx.x * 4 + threadIdx.y) * 16;
  if (m0 >= N) return;                         // wave-uniform: EXEC stays all-1s
  int half = lane >> 4;                        // 0 | 1
  int l = lane & 15;
  v8f acc0 = {}, acc1 = {}, acc2 = {}, acc3 = {};
  for (int k = 0; k < K; k += 4) {
    const float* ap = X + (size_t)(m0 + l) * K + (k + half * 2);
    v2f a;
    a.x = ap[0];
    a.y = ap[1];
    const float* bp = W + (size_t)(k + half * 2) * C + l;
    v2f b0, b1, b2, b3;
    b0.x = bp[0];
    b0.y = bp[C];
    b1.x = bp[16];
    b1.y = bp[C + 16];
    b2.x = bp[32];
    b2.y = bp[C + 32];
    b3.x = bp[48];
    b3.y = bp[C + 48];
    acc0 = __builtin_amdgcn_wmma_f32_16x16x4_f32(false, a, false, b0, (short)0,
                                                 acc0, false, false);
    acc1 = __builtin_amdgcn_wmma_f32_16x16x4_f32(false, a, false, b1, (short)0,
                                                 acc1, false, false);
    acc2 = __builtin_amdgcn_wmma_f32_16x16x4_f32(false, a, false, b2, (short)0,
                                                 acc2, false, false);
    acc3 = __builtin_amdgcn_wmma_f32_16x16x4_f32(false, a, false, b3, (short)0,
                                                 acc3, false, false);
  }
  float bv0 = bias[l];
  float bv1 = bias[16 + l];
  float bv2 = bias[32 + l];
  float bv3 = bias[48 + l];
  float* yp = Y + (size_t)m0 * C + l;
#pragma unroll
  for (int r = 0; r < 8; ++r) {
    size_t ro = (size_t)(r + half * 8) * C;
    yp[ro + 0] = acc0[r] + bv0;
    yp[ro + 16] = acc1[r] + bv1;
    yp[ro + 32] = acc2[r] + bv2;
    yp[ro + 48] = acc3[r] + bv3;
  }
}

// -------------------- BatchNorm (training stats) ----------------------------
// pass 1: per-column sum / sumsq. block = C threads, each block covers ROWS rows.
#define BN_ROWS 256
__global__ void k_bn_stats(const float* __restrict__ Y, float* __restrict__ st,
                           int N, int C) {
  int c = threadIdx.x;
  int r0 = blockIdx.x * BN_ROWS;
  int rend = min(r0 + BN_ROWS, N);
  float s = 0.0f, s2 = 0.0f;
  for (int r = r0; r < rend; ++r) {
    float v = Y[(size_t)r * C + c];
    s += v;
    s2 += v * v;
  }
  atomAddF32(&st[c], s);
  atomAddF32(&st[C + c], s2);
}

// pass 2: y = relu((y-mean)*rsqrt(var+eps)*g + bt)
__global__ void k_bn_apply_relu(float* __restrict__ Y,
                                const float* __restrict__ st,
                                const float* __restrict__ g,
                                const float* __restrict__ bt, int N, int C) {
  long i = (long)blockIdx.x * blockDim.x + threadIdx.x;
  if (i >= (long)N * C) return;
  int c = (int)(i % C);
  float invN = 1.0f / (float)N;
  float mean = st[c] * invN;
  float var = st[C + c] * invN - mean * mean;
  float v = (Y[i] - mean) * rsqrtf(var + BN_EPS) * g[c] + bt[c];
  Y[i] = fmaxf(v, 0.0f);
}

// ----------------------- GINE pre/post elementwise --------------------------
__global__ void k_combine(const float* __restrict__ h,
                          const float* __restrict__ agg,
                          const float* __restrict__ eps,
                          float* __restrict__ out, long n) {
  long i = (long)blockIdx.x * blockDim.x + threadIdx.x;
  if (i < n) out[i] = (1.0f + eps[0]) * h[i] + agg[i];
}

__global__ void k_relu_skip(const float* __restrict__ y,
                            const float* __restrict__ hprev,
                            float* __restrict__ out, long n, int use_skip) {
  long i = (long)blockIdx.x * blockDim.x + threadIdx.x;
  if (i >= n) return;
  float v = fmaxf(y[i], 0.0f);
  out[i] = use_skip ? (v + hprev[i]) : v;
}

// ------------------- segmented pooling (batch is sorted) --------------------
#define POOL_ROWS 128
__global__ void k_pool(const float* __restrict__ rep,
                       const int* __restrict__ batch,
                       float* __restrict__ pooled, int N, int C) {
  int c = threadIdx.x;               // one channel per thread
  int r0 = blockIdx.x * POOL_ROWS;
  if (r0 >= N) return;
  int rend = min(r0 + POOL_ROWS, N);
  int g = batch[r0];
  float acc = 0.0f;
  for (int r = r0; r < rend; ++r) {
    int bg = batch[r];
    if (bg != g) {                   // graph boundary: flush (rare)
      atomAddF32(&pooled[g * C + c], acc);
      acc = 0.0f;
      g = bg;
    }
    acc += rep[(size_t)r * C + c];
  }
  atomAddF32(&pooled[g * C + c], acc);
}

// -------------- final readout layer: [64,64]@[64,4]+b, out += ---------------
__global__ void k_readout_final(const float* __restrict__ H,
                                const float* __restrict__ W,
                                const float* __restrict__ b,
                                float* __restrict__ out) {
  int t = threadIdx.x;               // 256 threads: 64 graphs x 4 outputs
  int row = t >> 2, col = t & 3;
  float s = b[col];
#pragma unroll 8
  for (int k = 0; k < HID; ++k) s += H[row * HID + k] * W[k * OUT_CH + col];
  out[row * OUT_CH + col] += s;      // each thread owns one element
}

// ---------------------------------------------------------------------------
extern "C" void kernel_launch(void* const* d_in, const int* in_sizes, int n_in,
                              void* d_out, int out_size, void* d_ws,
                              size_t ws_size, hipStream_t stream) {
  const float* x = (const float*)d_in[0];
  const int* ei = (const int*)d_in[1];
  const float* ea = (const float*)d_in[2];
  const int* batch = (const int*)d_in[3];
  const int* src = ei;
  const int* dst = ei + N_EDGES;

  // Param pointer table (jax tree_flatten, sorted dict keys):
  // conv i: We, be, eps, W0,W1,W2, b0,b1,b2, bt0,bt1, g0,g1   (13 each)
  // lin  i: W0,W1,W2, b0,b1,b2, bt0,bt1, g0,g1                (10 each)
  const int CV = 4;            // conv base
  const int LN = 4 + 3 * 13;   // linear base = 43
  auto cvp = [&](int i, int j) { return (const float*)d_in[CV + 13 * i + j]; };
  auto lnp = [&](int i, int j) { return (const float*)d_in[LN + 10 * i + j]; };

  // workspace layout
  float* ws = (float*)d_ws;
  const size_t NB = (size_t)N_NODES * HID;   // 6.4M floats
  float* agg = ws;                           // [N, 64] (use first N*C)
  float* tA = ws + NB;
  float* tB = ws + 2 * NB;
  float* hA = ws + 3 * NB;
  float* hB = ws + 4 * NB;
  float* sm = ws + 5 * NB;
  float* stats = sm;                         // [128]
  float* pooled0 = sm + 256;                 // [64,16] (alloc 64*64 each)
  float* pooled1 = pooled0 + N_GRAPHS * HID;
  float* pooled2 = pooled1 + N_GRAPHS * HID;
  float* pooled3 = pooled2 + N_GRAPHS * HID;
  float* q1 = pooled3 + N_GRAPHS * HID;      // [64,64]
  float* q2 = q1 + N_GRAPHS * HID;

  auto zero = [&](float* p, long n) {
    k_zero<<<dim3((unsigned)((n + 255) / 256)), 256, 0, stream>>>(p, n);
  };
  auto gemm = [&](const float* X, const float* W, const float* b, float* Y,
                  int N, int K) {
    dim3 blk(32, 4);
    dim3 grd((N / 16 + 3) / 4);
    k_wmma_gemm64<<<grd, blk, 0, stream>>>(X, W, b, Y, N, K);
  };
  auto bn_relu = [&](float* Y, const float* g, const float* bt, int N, int C) {
    zero(stats, 2 * C);
    k_bn_stats<<<dim3((N + BN_ROWS - 1) / BN_ROWS), C, 0, stream>>>(Y, stats, N, C);
    long n = (long)N * C;
    k_bn_apply_relu<<<dim3((unsigned)((n + 255) / 256)), 256, 0, stream>>>(
        Y, stats, g, bt, N, C);
  };
  auto pool = [&](const float* rep, float* dstp, int C) {
    k_pool<<<dim3((N_NODES + POOL_ROWS - 1) / POOL_ROWS), C, 0, stream>>>(
        rep, batch, dstp, N_NODES, C);
  };

  // zero accumulators (ws/d_out are poisoned by harness)
  zero((float*)d_out, N_GRAPHS * OUT_CH);
  zero(pooled0, 4 * N_GRAPHS * HID);

  // ---- readout rep 0: raw x ----
  pool(x, pooled0, IN_CH);

  // ---- three GINE convs ----
  const float* hcur = x;
  float* reps[3] = {hA, hB, hA};   // ping-pong; skip input stays live
  for (int i = 0; i < 3; ++i) {
    int cin = (i == 0) ? IN_CH : HID;
    long ncin = (long)N_NODES * cin;
    zero(agg, ncin);
    k_edge_scatter<<<dim3((N_EDGES + 255) / 256, cin / 4), 256, 0, stream>>>(
        hcur, src, dst, ea, cvp(i, 0), cvp(i, 1), agg, N_EDGES, cin);
    // t = (1+eps)*h + agg  -> tB
    k_combine<<<dim3((unsigned)((ncin + 255) / 256)), 256, 0, stream>>>(
        hcur, agg, cvp(i, 2), tB, ncin);
    // MLP: [cin,64,64,64] with BN+ReLU on first two layers
    gemm(tB, cvp(i, 3), cvp(i, 6), tA, N_NODES, cin);
    bn_relu(tA, cvp(i, 11), cvp(i, 9), N_NODES, HID);
    gemm(tA, cvp(i, 4), cvp(i, 7), tB, N_NODES, HID);
    bn_relu(tB, cvp(i, 12), cvp(i, 10), N_NODES, HID);
    gemm(tB, cvp(i, 5), cvp(i, 8), tA, N_NODES, HID);
    // h_new = relu(mlp) (+ h_prev for convs 1,2)
    k_relu_skip<<<dim3((unsigned)((NB + 255) / 256)), 256, 0, stream>>>(
        tA, hcur, reps[i], (long)NB, i > 0 ? 1 : 0);
    hcur = reps[i];
    pool(hcur, (i == 0) ? pooled1 : (i == 1) ? pooled2 : pooled3, HID);
  }

  // ---- readout MLPs on pooled reps, accumulate into d_out ----
  float* pooled[4] = {pooled0, pooled1, pooled2, pooled3};
  for (int r = 0; r < 4; ++r) {
    int cin = (r == 0) ? IN_CH : HID;
    gemm(pooled[r], lnp(r, 0), lnp(r, 3), q1, N_GRAPHS, cin);
    bn_relu(q1, lnp(r, 8), lnp(r, 6), N_GRAPHS, HID);
    gemm(q1, lnp(r, 1), lnp(r, 4), q2, N_GRAPHS, HID);
    bn_relu(q2, lnp(r, 9), lnp(r, 7), N_GRAPHS, HID);
    k_readout_final<<<1, 256, 0, stream>>>(q2, lnp(r, 2), lnp(r, 5),
                                           (float*)d_out);
  }
}